// HierarchicalPathNetworkLayer_57758720196984
// MI455X (gfx1250) — compile-verified
//
#include <hip/hip_runtime.h>
#include <hip/hip_bf16.h>

typedef __attribute__((ext_vector_type(2))) float v2f;
typedef __attribute__((ext_vector_type(8))) float v8f;

#define N1_ 100000
#define N2_ 200000
#define F_  128

// ---------------- upward scatter: h2[dst] += feat[src] ----------------
__global__ void up_pass_kernel(const int* __restrict__ src12,
                               const int* __restrict__ dst12,
                               const float* __restrict__ feat,
                               float* __restrict__ h2,
                               int nEdges) {
    long long t = (long long)blockIdx.x * blockDim.x + threadIdx.x;
    int e = (int)(t >> 7);
    int f = (int)(t & 127);
    if (e >= nEdges) return;
    int s = src12[e];
    int d = dst12[e];
    float v = feat[(size_t)s * F_ + f];
    unsafeAtomicAdd(&h2[(size_t)d * F_ + f], v);
}

// ---------------- per-row softmax stats (max, 1/sumexp), one wave32/row ----------------
__global__ void softmax_stats_kernel(const float* __restrict__ h2,
                                     float* __restrict__ rowmax,
                                     float* __restrict__ rowinv,
                                     int nRows) {
    int wave = (blockIdx.x * blockDim.x + threadIdx.x) >> 5;
    int lane = threadIdx.x & 31;
    if (wave >= nRows) return;
    const float4* r = (const float4*)(h2 + (size_t)wave * F_);
    float4 v = r[lane];
    float m = fmaxf(fmaxf(v.x, v.y), fmaxf(v.z, v.w));
    #pragma unroll
    for (int o = 16; o >= 1; o >>= 1) m = fmaxf(m, __shfl_xor(m, o, 32));
    float s = __expf(v.x - m) + __expf(v.y - m) + __expf(v.z - m) + __expf(v.w - m);
    #pragma unroll
    for (int o = 16; o >= 1; o >>= 1) s += __shfl_xor(s, o, 32);
    if (lane == 0) {
        rowmax[wave] = m;
        rowinv[wave] = 1.0f / s;
    }
}

// ------------- fused downward pass: h2_agg[src]+=h2[dst]; h_down[src]+=softmax(h2)[dst] -------------
__global__ void down_pass_kernel(const int* __restrict__ src12,
                                 const int* __restrict__ dst12,
                                 const float* __restrict__ h2,
                                 const float* __restrict__ rowmax,
                                 const float* __restrict__ rowinv,
                                 float* __restrict__ h2agg,
                                 float* __restrict__ hdown,
                                 int nEdges) {
    long long t = (long long)blockIdx.x * blockDim.x + threadIdx.x;
    int e = (int)(t >> 7);
    int f = (int)(t & 127);
    if (e >= nEdges) return;
    int s = src12[e];
    int d = dst12[e];
    float v = h2[(size_t)d * F_ + f];
    unsafeAtomicAdd(&h2agg[(size_t)s * F_ + f], v);
    float sm = __expf(v - rowmax[d]) * rowinv[d];
    unsafeAtomicAdd(&hdown[(size_t)s * F_ + f], sm);
}

// ------------- GEMM + bias + SiLU via V_WMMA_F32_16X16X4_F32 -------------
// out[M,128] = silu( [h1 | h2agg | hdown] @ W + b ),  W is [384,128] row-major.
// One wave per 16-row M tile; 8 f32 accumulators cover the full N=128.
__global__ void gemm_silu_kernel(const float* __restrict__ h1,
                                 const float* __restrict__ h2agg,
                                 const float* __restrict__ hdown,
                                 const float* __restrict__ W,
                                 const float* __restrict__ bias,
                                 float* __restrict__ out,
                                 int M) {
    const int lane = threadIdx.x & 31;
    const int mt   = blockIdx.x * (blockDim.x >> 5) + (threadIdx.x >> 5);
    if (mt * 16 >= M) return;              // wave-uniform: EXEC all-ones inside

    const int half = lane >> 4;            // selects K pair (A) / K pair (B)
    const int l    = lane & 15;            // M row (A) / N col (B)
    const int row  = mt * 16 + l;

    v8f acc[8];
    #pragma unroll
    for (int i = 0; i < 8; ++i) acc[i] = (v8f){0.f,0.f,0.f,0.f,0.f,0.f,0.f,0.f};

    const float* srcs[3] = {h1, h2agg, hdown};
    #pragma unroll
    for (int blk = 0; blk < 3; ++blk) {
        const float* xr = srcs[blk] + (size_t)row * F_;
        const float* wb = W + (size_t)blk * F_ * F_;
        for (int k = 0; k < F_; k += 4) {
            const int ka = k + half * 2;
            // A fragment 16x4 f32: lanes 0-15 -> K=ka,ka+1 ; lanes 16-31 -> K=k+2,k+3
            v2f a;
            a.x = xr[ka];
            a.y = xr[ka + 1];
            const float* w0 = wb + (size_t)ka * F_ + l;
            #pragma unroll
            for (int nt = 0; nt < 8; ++nt) {
                // B fragment 4x16 f32: lane = N, VGPR0/1 = K pair within lane-half
                v2f bfr;
                bfr.x = w0[nt * 16];
                bfr.y = w0[nt * 16 + F_];
                acc[nt] = __builtin_amdgcn_wmma_f32_16x16x4_f32(
                    /*neg_a=*/false, a, /*neg_b=*/false, bfr,
                    /*c_mod=*/(short)0, acc[nt],
                    /*reuse_a=*/false, /*reuse_b=*/false);
            }
        }
    }

    // Epilogue: C/D layout — VGPR r: lanes 0-15 -> (M=r, N=l), lanes 16-31 -> (M=8+r, N=l)
    #pragma unroll
    for (int nt = 0; nt < 8; ++nt) {
        const float bv = bias[nt * 16 + l];
        #pragma unroll
        for (int r = 0; r < 8; ++r) {
            const int m = mt * 16 + half * 8 + r;
            float v = acc[nt][r] + bv;
            out[(size_t)m * F_ + nt * 16 + l] = v / (1.0f + __expf(-v));
        }
    }
}

extern "C" void kernel_launch(void* const* d_in, const int* in_sizes, int n_in,
                              void* d_out, int out_size, void* d_ws, size_t ws_size,
                              hipStream_t stream) {
    const float* feat  = (const float*)d_in[0];
    const float* W     = (const float*)d_in[1];
    const float* bias  = (const float*)d_in[2];
    const int*   src12 = (const int*)d_in[3];
    const int*   dst12 = (const int*)d_in[4];
    // d_in[5..8] (levels 3/4 edges) are dead w.r.t. the reference output.

    const int nEdges = in_sizes[3];        // 2*N2 = 400000
    const int M      = in_sizes[0] / F_;   // N1 = 100000

    // workspace layout (bytes)
    char* ws = (char*)d_ws;
    float* h2     = (float*)(ws);                                   // N2*128 f32
    float* h2agg  = (float*)(ws + (size_t)N2_ * F_ * 4);            // N1*128 f32
    float* hdown  = (float*)(ws + (size_t)(N2_ + N1_) * F_ * 4);    // N1*128 f32
    float* rowmax = (float*)(ws + (size_t)(N2_ + 2 * N1_) * F_ * 4);
    float* rowinv = (float*)(ws + (size_t)(N2_ + 2 * N1_) * F_ * 4 + (size_t)N2_ * 4);

    hipMemsetAsync(h2,    0, (size_t)N2_ * F_ * 4, stream);
    hipMemsetAsync(h2agg, 0, (size_t)N1_ * F_ * 4, stream);
    hipMemsetAsync(hdown, 0, (size_t)N1_ * F_ * 4, stream);

    {   // upward scatter
        long long threads = (long long)nEdges * F_;
        int blocks = (int)((threads + 255) / 256);
        up_pass_kernel<<<blocks, 256, 0, stream>>>(src12, dst12, feat, h2, nEdges);
    }
    {   // per-row softmax stats, one wave32 per row
        int waves_per_block = 256 / 32;
        int blocks = (N2_ + waves_per_block - 1) / waves_per_block;
        softmax_stats_kernel<<<blocks, 256, 0, stream>>>(h2, rowmax, rowinv, N2_);
    }
    {   // fused downward gather + scatter
        long long threads = (long long)nEdges * F_;
        int blocks = (int)((threads + 255) / 256);
        down_pass_kernel<<<blocks, 256, 0, stream>>>(src12, dst12, h2, rowmax, rowinv,
                                                     h2agg, hdown, nEdges);
    }
    {   // WMMA GEMM + SiLU
        int mtiles = (M + 15) / 16;            // 6250
        int waves_per_block = 128 / 32;        // 4
        int blocks = (mtiles + waves_per_block - 1) / waves_per_block;
        gemm_silu_kernel<<<blocks, 128, 0, stream>>>(feat, h2agg, hdown, W, bias,
                                                     (float*)d_out, M);
    }
}